// NodeClsPooler_2319282340537
// MI455X (gfx1250) — compile-verified
//
#include <hip/hip_runtime.h>

// ---------------------------------------------------------------------------
// NodeClsPooler: pooled = x[first_idx]; out = pooled @ W^T + b
//   B = 8192 graphs, C = 128 channels, f32 in / f32 out.
// Memory-bound (~8 MB HBM traffic) -> use f32 WMMA (exact precision, and the
// matrix op is free at this arithmetic intensity).
// ---------------------------------------------------------------------------

typedef float v2f __attribute__((ext_vector_type(2)));
typedef float v8f __attribute__((ext_vector_type(8)));

#define CHN    128
#define TILE_M 16    // graphs per block
#define TILE_N 64    // output columns per block (4 waves x 16)
#define APAD   132   // padded LDS row stride (floats): bank = 4*lane + c, conflict-free

__global__ __launch_bounds__(128)
void node_cls_pool_linear(const float* __restrict__ x,
                          const int*   __restrict__ batch,
                          const float* __restrict__ W,
                          const float* __restrict__ bias,
                          float*       __restrict__ out,
                          int nNodes)
{
    __shared__ float As[TILE_M][APAD];   // gathered pooled rows   (16 x 128)
    __shared__ float Ws[TILE_N][APAD];   // W rows for this column slab (64 x 128)
    __shared__ int   sFirst[TILE_M];

    const int tid     = threadIdx.x;
    const int g0      = blockIdx.y * TILE_M;   // first graph of this tile
    const int colBase = blockIdx.x * TILE_N;   // first output column of this tile

    // --- 1) first node index per graph: lower_bound over sorted batch -------
    if (tid < TILE_M) {
        const int g = g0 + tid;
        int lo = 0, hi = nNodes;
        while (lo < hi) {
            const int mid = (lo + hi) >> 1;
            if (batch[mid] < g) lo = mid + 1; else hi = mid;
        }
        sFirst[tid] = lo;
    }
    __syncthreads();

    // --- 2) stage A tile (gathered rows) and W slab into LDS, float4 loads --
    for (int idx = tid; idx < TILE_M * (CHN / 4); idx += blockDim.x) {
        const int row = idx >> 5;
        const int c4  = (idx & 31) << 2;
        const float4 v = *(const float4*)&x[(long long)sFirst[row] * CHN + c4];
        *(float4*)&As[row][c4] = v;
    }
    for (int idx = tid; idx < TILE_N * (CHN / 4); idx += blockDim.x) {
        const int row = idx >> 5;
        const int c4  = (idx & 31) << 2;
        const float4 v = *(const float4*)&W[(long long)(colBase + row) * CHN + c4];
        *(float4*)&Ws[row][c4] = v;
    }
    __syncthreads();

    // --- 3) WMMA f32 16x16x4: each wave owns one 16x16 output tile ----------
    const int wave = tid >> 5;
    const int lane = tid & 31;
    const int l16  = lane & 15;
    const int lhi  = lane >> 4;               // 0: K=k0..k0+1 ; 1: K=k0+2..k0+3
    const int wrow = wave * 16 + l16;         // row of Ws this lane streams (N = l16)

    v8f acc = {0.f, 0.f, 0.f, 0.f, 0.f, 0.f, 0.f, 0.f};

    #pragma unroll 4
    for (int k0 = 0; k0 < CHN; k0 += 4) {
        const int ka = k0 + 2 * lhi;
        // A 16x4 (MxK) f32: lane l16 holds row M=l16, K-pair per half-wave
        const v2f a = *(const v2f*)&As[l16][ka];
        // B 4x16 (KxN) f32: B[k][n] = W[colBase + wave*16 + n][k0 + k]
        const v2f b = *(const v2f*)&Ws[wrow][ka];
        acc = __builtin_amdgcn_wmma_f32_16x16x4_f32(
                  /*neg_a=*/false, a, /*neg_b=*/false, b,
                  /*c_mod=*/(short)0, acc, /*reuse_a=*/false, /*reuse_b=*/false);
    }

    // --- 4) add bias, store. C/D layout: (VGPR r, lane) -> M = r + 8*lhi, N = l16
    const int   j  = colBase + wave * 16 + l16;
    const float bv = bias[j];
    #pragma unroll
    for (int r = 0; r < 8; ++r) {
        const int m = r + 8 * lhi;
        out[(long long)(g0 + m) * CHN + j] = acc[r] + bv;
    }
}

extern "C" void kernel_launch(void* const* d_in, const int* in_sizes, int n_in,
                              void* d_out, int out_size, void* d_ws, size_t ws_size,
                              hipStream_t stream)
{
    const float* x     = (const float*)d_in[0];
    const int*   batch = (const int*)  d_in[1];
    const float* W     = (const float*)d_in[2];
    const float* b     = (const float*)d_in[3];
    float*       out   = (float*)      d_out;

    const int nNodes    = in_sizes[1];        // TOTAL_NODES
    const int numGraphs = out_size / CHN;     // 8192

    dim3 grid(CHN / TILE_N, numGraphs / TILE_M);   // (2, 512)
    node_cls_pool_linear<<<grid, 128, 0, stream>>>(x, batch, W, b, out, nNodes);
}